// FusedEmbedding_49701361549967
// MI455X (gfx1250) — compile-verified
//
#include <hip/hip_runtime.h>

typedef float v2f __attribute__((ext_vector_type(2)));
typedef float v8f __attribute__((ext_vector_type(8)));

#define SMALL 512
#define LARGE 1024
#define NPOS  8192                 // B*S = 4*2048
#define ROWS  32                   // token rows per workgroup (M-block)
#define STR   (SMALL + 4)          // 516: row stride % 64 banks == 4 -> conflict-free

__global__ __launch_bounds__(256)
void fused_embed_wmma(const int* __restrict__ ids,
                      const float* __restrict__ weight,    // (VOCAB, SMALL)
                      const float* __restrict__ cw,        // (LARGE-SMALL, SMALL) row-major
                      const float* __restrict__ residual,  // (VOCAB, LARGE)
                      float* __restrict__ out)             // (NPOS, LARGE)
{
    __shared__ __align__(16) float As[ROWS * STR];         // 66,048 B (CDNA5 LDS up to 320KB/WG)

    const unsigned pos0 = blockIdx.x * ROWS;
    const unsigned tid  = threadIdx.x;

    // ---- Phase 1: gather 32 weight rows -> LDS; emit out[:, 0:SMALL] = w + residual ----
    {
        const unsigned row0 = tid >> 4;   // 0..15
        const unsigned c16  = tid & 15;   // 0..15 (float4 slot)
#pragma unroll
        for (unsigned rr = 0; rr < 2; ++rr) {
            const unsigned row = row0 + rr * 16;
            const unsigned t   = (unsigned)ids[pos0 + row];
            const unsigned w4  = t << 7;                 // weight row in float4 units (512/4)
            const unsigned r4  = t << 8;                 // residual row in float4 units (1024/4)
            const unsigned o4  = (pos0 + row) << 8;      // out row in float4 units
            const float4* wv = (const float4*)weight;
            const float4* rv = (const float4*)residual;
            float4*       ov = (float4*)out;
            float*        lrow = As + row * STR;
#pragma unroll
            for (unsigned j = 0; j < 8; ++j) {
                const unsigned c4 = c16 + j * 16;        // 0..127 float4s per row
                float4 w = wv[w4 + c4];
                float4 r = rv[r4 + c4];
                *(float4*)(lrow + c4 * 4) = w;
                float4 o;
                o.x = w.x + r.x; o.y = w.y + r.y; o.z = w.z + r.z; o.w = w.w + r.w;
                ov[o4 + c4] = o;
            }
        }
    }
    __syncthreads();

    // ---- Phase 2: expanded(32 x 512) = As @ cw^T, 2 M-tiles x 4 N-tiles per wave ----
    const unsigned wave   = tid >> 5;     // 0..7
    const unsigned lane   = tid & 31;
    const unsigned lane16 = lane & 15;
    const unsigned hi     = lane >> 4;    // 0 or 1
    const unsigned koff   = hi * 2;       // K sub-offset per ISA 16x4 / 4x16 f32 layout

    const float* a0p = As + lane16 * STR + koff;          // M-tile 0: rows 0..15
    const float* a1p = As + (16 + lane16) * STR + koff;   // M-tile 1: rows 16..31

    const unsigned nb0 = wave * 64;       // this wave covers expanded cols [nb0, nb0+64)
    const float* b0p = cw + (((nb0 +      lane16) << 9) + koff);
    const float* b1p = cw + (((nb0 + 16 + lane16) << 9) + koff);
    const float* b2p = cw + (((nb0 + 32 + lane16) << 9) + koff);
    const float* b3p = cw + (((nb0 + 48 + lane16) << 9) + koff);

    v8f c00 = {}, c01 = {}, c02 = {}, c03 = {};   // M-tile 0
    v8f c10 = {}, c11 = {}, c12 = {}, c13 = {};   // M-tile 1

#pragma unroll 2
    for (unsigned k = 0; k < SMALL; k += 4) {
        v2f a0 = *(const v2f*)(a0p + k);
        v2f a1 = *(const v2f*)(a1p + k);
        v2f b0 = *(const v2f*)(b0p + k);
        v2f b1 = *(const v2f*)(b1p + k);
        v2f b2 = *(const v2f*)(b2p + k);
        v2f b3 = *(const v2f*)(b3p + k);
        c00 = __builtin_amdgcn_wmma_f32_16x16x4_f32(false, a0, false, b0, (short)0, c00, false, false);
        c10 = __builtin_amdgcn_wmma_f32_16x16x4_f32(false, a1, false, b0, (short)0, c10, false, false);
        c01 = __builtin_amdgcn_wmma_f32_16x16x4_f32(false, a0, false, b1, (short)0, c01, false, false);
        c11 = __builtin_amdgcn_wmma_f32_16x16x4_f32(false, a1, false, b1, (short)0, c11, false, false);
        c02 = __builtin_amdgcn_wmma_f32_16x16x4_f32(false, a0, false, b2, (short)0, c02, false, false);
        c12 = __builtin_amdgcn_wmma_f32_16x16x4_f32(false, a1, false, b2, (short)0, c12, false, false);
        c03 = __builtin_amdgcn_wmma_f32_16x16x4_f32(false, a0, false, b3, (short)0, c03, false, false);
        c13 = __builtin_amdgcn_wmma_f32_16x16x4_f32(false, a1, false, b3, (short)0, c13, false, false);
    }

    // ---- Epilogue: out[:, SMALL:] = expanded + residual[:, SMALL:] ----
    // C/D layout: VGPR v -> M = v + 8*hi (within tile); N = lane16
    int toks0[8], toks1[8];
#pragma unroll
    for (unsigned v = 0; v < 8; ++v) {
        toks0[v] = ids[pos0 + hi * 8 + v];
        toks1[v] = ids[pos0 + 16 + hi * 8 + v];
    }

    const unsigned col = SMALL + nb0 + lane16;
#pragma unroll
    for (unsigned v = 0; v < 8; ++v) {
        const unsigned m0 = hi * 8 + v;
        const unsigned m1 = 16 + m0;
        const unsigned o0 = ((pos0 + m0) << 10) + col;           // 32-bit offsets:
        const unsigned o1 = ((pos0 + m1) << 10) + col;           //  max ~8.4M (out)
        const unsigned r0 = (((unsigned)toks0[v]) << 10) + col;  //  max ~51.5M (residual)
        const unsigned r1 = (((unsigned)toks1[v]) << 10) + col;
        out[o0]      = c00[v] + residual[r0];
        out[o0 + 16] = c01[v] + residual[r0 + 16];
        out[o0 + 32] = c02[v] + residual[r0 + 32];
        out[o0 + 48] = c03[v] + residual[r0 + 48];
        out[o1]      = c10[v] + residual[r1];
        out[o1 + 16] = c11[v] + residual[r1 + 16];
        out[o1 + 32] = c12[v] + residual[r1 + 32];
        out[o1 + 48] = c13[v] + residual[r1 + 48];
    }
}

extern "C" void kernel_launch(void* const* d_in, const int* in_sizes, int n_in,
                              void* d_out, int out_size, void* d_ws, size_t ws_size,
                              hipStream_t stream) {
    const int*   ids      = (const int*)d_in[0];
    const float* weight   = (const float*)d_in[1];
    const float* cw       = (const float*)d_in[2];
    const float* residual = (const float*)d_in[3];
    float*       out      = (float*)d_out;

    dim3 grid(NPOS / ROWS);          // 256 workgroups -> fills 128-256 WGPs
    dim3 block(256);                 // 8 wave32s
    fused_embed_wmma<<<grid, block, 0, stream>>>(ids, weight, cw, residual, out);
}